// NSGT_sliced_14602888806528
// MI455X (gfx1250) — compile-verified
//
#include <hip/hip_runtime.h>
#include <hip/hip_bf16.h>

// ---------------------------------------------------------------------------
// NSGT sliced transform for MI455X (gfx1250).
//
//   K1: per-frame window + 16384-pt complex FFT in LDS (data 128 KB + twiddle
//       table 64 KB = 192 KB dynamic LDS; CDNA5 WGP has 320 KB). One block
//       per (c,s) frame. ft (28.3 MB) -> d_ws, stays L2-resident (192 MB L2).
//   KE: length-M IDFT matrix in QUAD-PACKED layout:
//         Bpk[(k/2)*M+n] = {cos(k,n), cos(k+1,n), sin(k,n), sin(k+1,n)}/M
//       One global_load_b128 per lane yields BOTH B fragments (br = elems
//       0-1, bi = elems 2-3; each an even-aligned VGPR pair, legal WMMA src).
//       6.2 MB, L2-resident, reused by all row-blocks of K2.
//   K2: gather+IDFT as complex GEMM with V_WMMA_F32_16X16X4_F32 (fp32 keeps
//       transform accuracy). Each block owns 16 rows (16 (c,s,f) bins):
//       gathers its G tile into LDS ONCE in the same quad-packed layout
//       (16*(M/2) float4 = 112.6 KB), then 8 waves sweep the 55 column tiles.
//       Inner loop: 1 ds_load_b128 + 1 global_load_b128 + 4 WMMA
//       (RR,II,RI,IR; fp32 WMMA has no A/B negate -> combine at end) with
//       bumped-pointer addressing. Parity roll + (C,S,F,M)->(S,C,F,M)
//       transpose folded into per-lane precomputed store addresses.
//       Only the 374 MB output touches HBM (~16 us floor at 23.3 TB/s).
// ---------------------------------------------------------------------------

typedef float v2f __attribute__((ext_vector_type(2)));
typedef float v8f __attribute__((ext_vector_type(8)));

extern __shared__ char dynraw[];    // dynamic LDS, cast per kernel

// ---------------- Kernel 1: window + 16384-pt FFT (radix-2 DIT in LDS) -----
__global__ __launch_bounds__(512)
void nsgt_fft_frames(const float* __restrict__ x, const float* __restrict__ swin,
                     float2* __restrict__ ft, int N, int logN, int hop,
                     int S, int Lp) {
    const int frame = blockIdx.x;       // frame = c*S + s
    const int s     = frame % S;
    const int c     = frame / S;
    const size_t base = (size_t)c * (size_t)Lp + (size_t)s * (size_t)hop;
    float2* a = (float2*)dynraw;        // N complex
    float2* w = a + N;                  // N/2 twiddles: w[j] = e^{-2pi i j/N}

    // Twiddle table (once) + bit-reversed windowed load
    for (int j = threadIdx.x; j < (N >> 1); j += blockDim.x) {
        float sn, cs;
        sincosf(-6.28318530717958647692f * (float)j / (float)N, &sn, &cs);
        w[j] = make_float2(cs, sn);
    }
    for (int t = threadIdx.x; t < N; t += blockDim.x) {
        unsigned rev = __brev((unsigned)t) >> (32 - logN);
        float v = x[base + rev] * swin[rev];
        a[t] = make_float2(v, 0.0f);
    }
    __syncthreads();

    const int half_total = N >> 1;
    for (int st = 1; st <= logN; ++st) {
        const int half = 1 << (st - 1);
        const int tsh  = logN - st;     // twiddle(j,m) = w[j << tsh]
        for (int k = threadIdx.x; k < half_total; k += blockDim.x) {
            const int j  = k & (half - 1);
            const int i0 = ((k >> (st - 1)) << st) + j;
            const int i1 = i0 + half;
            const float2 tw = w[j << tsh];
            const float2 u = a[i0];
            const float2 t = a[i1];
            const float vr = t.x * tw.x - t.y * tw.y;
            const float vi = t.x * tw.y + t.y * tw.x;
            a[i0] = make_float2(u.x + vr, u.y + vi);
            a[i1] = make_float2(u.x - vr, u.y - vi);
        }
        __syncthreads();
    }

    float2* dst = ft + (size_t)frame * (size_t)N;
    for (int t = threadIdx.x; t < N; t += blockDim.x) dst[t] = a[t];
}

// ---------------- Kernel E: quad-packed IDFT matrix ------------------------
__global__ __launch_bounds__(256)
void nsgt_build_idft(float4* __restrict__ Bpk, int M) {
    const int i = blockIdx.x * blockDim.x + threadIdx.x;
    if (i >= (M >> 1) * M) return;
    const int kp = i / M;               // k pair
    const int n  = i - kp * M;
    const int k0 = kp << 1;
    const int t0 = (int)(((long long)k0 * n) % (long long)M);        // exact
    const int t1 = (int)(((long long)(k0 + 1) * n) % (long long)M);  // reduction
    const float inv = 1.0f / (float)M;
    float s0, c0, s1, c1;
    sincosf(6.28318530717958647692f * (float)t0 / (float)M, &s0, &c0);
    sincosf(6.28318530717958647692f * (float)t1 / (float)M, &s1, &c1);
    Bpk[i] = make_float4(c0 * inv, c1 * inv, s0 * inv, s1 * inv);
}

// ---------------- Kernel 2: gather + IDFT GEMM via fp32 WMMA ---------------
__global__ __launch_bounds__(256)
void nsgt_gemm_idft(const float2* __restrict__ ft, const int* __restrict__ idx,
                    const float* __restrict__ wmat,
                    const float4* __restrict__ Bpk,
                    float2* __restrict__ out, int S, int F, int M, int N) {
    const int C  = 2;
    const int R  = C * S * F;
    const int Mh = M >> 1;
    const int rowBase = blockIdx.x * 16;
    float4* Gpk = (float4*)dynraw;      // [16][M/2] {gr0,gr1,gi0,gi1}

    // ---- Phase A: gather 16 rows of G into quad-packed LDS (once) ----
    for (int e = threadIdx.x; e < 16 * Mh; e += blockDim.x) {
        const int r  = e / Mh;
        const int kp = e - r * Mh;
        const int m0 = kp << 1;
        const int row = rowBase + r;
        float4 g = make_float4(0.0f, 0.0f, 0.0f, 0.0f);
        if (row < R) {
            const int f  = row % F;
            const int cs = row / F;             // = c*S + s
            const int fb = f * M + m0;
            const int j0 = idx[fb];
            const int j1 = idx[fb + 1];
            const float w0 = wmat[fb];
            const float w1 = wmat[fb + 1];
            const float2 v0 = ft[(size_t)cs * (size_t)N + (size_t)j0];
            const float2 v1 = ft[(size_t)cs * (size_t)N + (size_t)j1];
            g = make_float4(v0.x * w0, v1.x * w1, v0.y * w0, v1.y * w1);
        }
        Gpk[e] = g;
    }
    __syncthreads();

    const int lane    = threadIdx.x & 31;
    const int wave    = threadIdx.x >> 5;
    const int nWaves  = blockDim.x >> 5;
    const int laneRow = lane & 15;          // A: M index / B,C,D: N index
    const int hiHalf  = lane >> 4;          // lanes 16-31 take K+2,K+3
    const int nCT     = M >> 4;             // column tiles
    const int quarter = M >> 2;
    const int kSteps  = M >> 2;             // K in chunks of 4

    // ---- Per-lane output decode, hoisted out of the tile loop ----
    // C/D layout: VGPR v -> (M=v, N=lane&15) lanes 0-15 ; (M=v+8, ...) 16-31
    const int mHi = hiHalf << 3;
    size_t outBase[8];
    int    shiftArr[8];
    bool   validArr[8];
#pragma unroll
    for (int v = 0; v < 8; ++v) {
        const int row = rowBase + v + mHi;
        validArr[v] = (row < R);
        if (validArr[v]) {
            const int f  = row % F;
            const int cs = row / F;
            const int s  = cs % S;
            const int c  = cs / S;
            shiftArr[v] = (s & 1) ? quarter : 3 * quarter;   // out[n']=c[(n'+sh)%M]
            outBase[v]  = (((size_t)s * C + c) * (size_t)F + f) * (size_t)M;
        } else { shiftArr[v] = 0; outBase[v] = 0; }
    }

    for (int ct = wave; ct < nCT; ct += nWaves) {
        const int col0 = ct << 4;
        const int col  = col0 + laneRow;
        v8f accRR = {}; v8f accII = {}; v8f accRI = {}; v8f accIR = {};

        // Bumped-pointer operand streams (constant strides, offsets fold)
        const float4* aP = Gpk + laneRow * Mh + hiHalf;               // +2/step
        const float4* bP = Bpk + (size_t)hiHalf * (size_t)M + (size_t)col;
        const size_t  bStride = 2 * (size_t)M;                        // /step

        for (int ks = 0; ks < kSteps; ++ks) {
            const float4 aq = *aP;      // {Gr[kk],Gr[kk+1],Gi[kk],Gi[kk+1]}
            const float4 bq = *bP;      // {Er[kk],Er[kk+1],Ei[kk],Ei[kk+1]}[col]
            const v2f ar = {aq.x, aq.y};
            const v2f ai = {aq.z, aq.w};
            const v2f br = {bq.x, bq.y};
            const v2f bi = {bq.z, bq.w};
            // Cr = RR - II ; Ci = RI + IR (combined after the loop)
            accRR = __builtin_amdgcn_wmma_f32_16x16x4_f32(false, ar, false, br,
                        (short)0, accRR, false, false);
            accII = __builtin_amdgcn_wmma_f32_16x16x4_f32(false, ai, false, bi,
                        (short)0, accII, false, false);
            accRI = __builtin_amdgcn_wmma_f32_16x16x4_f32(false, ar, false, bi,
                        (short)0, accRI, false, false);
            accIR = __builtin_amdgcn_wmma_f32_16x16x4_f32(false, ai, false, br,
                        (short)0, accIR, false, false);
            aP += 2; bP += bStride;
            // Speculative stream-ahead (3 K-steps); safe per ISA 10.5 --
            // invalid translations silently dropped.
            __builtin_prefetch((const void*)(bP + bStride * 3), 0, 1);
        }

        const int n = col0 + laneRow;
#pragma unroll
        for (int v = 0; v < 8; ++v) {
            if (validArr[v]) {
                int outN = n - shiftArr[v]; if (outN < 0) outN += M;
                const float cr = accRR[v] - accII[v];
                const float ci = accRI[v] + accIR[v];
                out[outBase[v] + (size_t)outN] = make_float2(cr, ci);
            }
        }
    }
}

// ---------------------------------------------------------------------------
extern "C" void kernel_launch(void* const* d_in, const int* in_sizes, int n_in,
                              void* d_out, int out_size, void* d_ws, size_t ws_size,
                              hipStream_t stream) {
    const float* x    = (const float*)d_in[0];
    const int*   idx  = (const int*)  d_in[1];
    const float* wmat = (const float*)d_in[2];
    const float* swin = (const float*)d_in[3];

    const int C   = 2;                       // fixed in reference
    const int F   = 246;                     // fixed filterbank: 1+122+1+122
    const int N   = in_sizes[3];             // SL_LEN = 16384
    const int Lp  = in_sizes[0] / C;
    const int hop = N / 2;
    const int S   = (Lp - N) / hop + 1;      // 108
    const int M   = in_sizes[1] / F;         // 880
    int logN = 0; while ((1 << logN) < N) ++logN;

    // Workspace: [ft : C*S*N float2][Bpk : (M/2)*M float4]
    float2* ft = (float2*)d_ws;
    size_t off = ((size_t)C * S * (size_t)N * sizeof(float2) + 255u) & ~(size_t)255u;
    float4* Bpk = (float4*)((char*)d_ws + off);

    // K1: one block per frame; dynamic LDS = N complex + N/2 twiddles (192 KB)
    nsgt_fft_frames<<<C * S, 512, (size_t)N * sizeof(float2) * 3 / 2, stream>>>(
        x, swin, ft, N, logN, hop, S, Lp);

    // KE: quad-packed IDFT matrix
    const int tot = (M >> 1) * M;
    nsgt_build_idft<<<(tot + 255) / 256, 256, 0, stream>>>(Bpk, M);

    // K2: 16 rows per block, 8 waves, LDS = 16*(M/2) float4 (112.6 KB)
    const int R = C * S * F;
    const int nBlocks = (R + 15) / 16;
    nsgt_gemm_idft<<<nBlocks, 256, (size_t)16 * (M >> 1) * sizeof(float4), stream>>>(
        ft, idx, wmat, Bpk, (float2*)d_out, S, F, M, N);
}